// Repro_63428077027476
// MI455X (gfx1250) — compile-verified
//
#include <hip/hip_runtime.h>

#define N_NODES 50000
#define N_EDGES 1250000
#define D_IN    256
#define D_OUT   64

typedef __attribute__((ext_vector_type(2))) float v2f;
typedef __attribute__((ext_vector_type(8))) float v8f;

// ---------------------------------------------------------------------------
// Kernel 1: H = X @ W^T using V_WMMA_F32_16X16X4_F32 (full fp32 precision).
// One wave per 16-row tile of X; each wave produces a 16x64 tile of H using
// 4 accumulators (N split into 4 tiles of 16). K=256 consumed 4 at a time.
//
// A-frag (16x4 f32): lane m=lane&15 holds row m; kh=lane>>4 selects K-half;
//   VGPR0/1 = K=2*kh, 2*kh+1  -> one float2 load from X row.
// B-frag (4x16 f32): lane n=lane&15 holds col n; B[k][n] = W[n][k]
//   -> one float2 load from W row n at K=2*kh.
// C/D (16x16 f32): VGPR v -> row v+8*kh, col lane&15.
// ---------------------------------------------------------------------------
__global__ __launch_bounds__(256) void gemm_wmma_f32(
    const float* __restrict__ X,   // [N_NODES, D_IN]
    const float* __restrict__ W,   // [D_OUT, D_IN]
    float* __restrict__ H)         // [N_NODES, D_OUT]
{
    const int waveInBlock = threadIdx.x >> 5;          // wave-uniform
    const int tile = blockIdx.x * 8 + waveInBlock;     // 16-row tile index
    if (tile * 16 >= N_NODES) return;                  // whole-wave exit: EXEC stays all-1s

    const int lane = threadIdx.x & 31;
    const int m  = lane & 15;    // A row / B col / C col
    const int kh = lane >> 4;    // K-half selector
    const int r0 = tile * 16;

    v8f acc0 = {}, acc1 = {}, acc2 = {}, acc3 = {};

    const float* xrow = X + (size_t)(r0 + m) * D_IN + 2 * kh;
    const float* w0   = W + (size_t)( 0 + m) * D_IN + 2 * kh;
    const float* w1   = W + (size_t)(16 + m) * D_IN + 2 * kh;
    const float* w2   = W + (size_t)(32 + m) * D_IN + 2 * kh;
    const float* w3   = W + (size_t)(48 + m) * D_IN + 2 * kh;

    for (int kb = 0; kb < D_IN; kb += 4) {
        v2f a  = *(const v2f*)(xrow + kb);
        v2f b0 = *(const v2f*)(w0 + kb);
        v2f b1 = *(const v2f*)(w1 + kb);
        v2f b2 = *(const v2f*)(w2 + kb);
        v2f b3 = *(const v2f*)(w3 + kb);
        acc0 = __builtin_amdgcn_wmma_f32_16x16x4_f32(false, a, false, b0, (short)0, acc0, false, false);
        acc1 = __builtin_amdgcn_wmma_f32_16x16x4_f32(false, a, false, b1, (short)0, acc1, false, false);
        acc2 = __builtin_amdgcn_wmma_f32_16x16x4_f32(false, a, false, b2, (short)0, acc2, false, false);
        acc3 = __builtin_amdgcn_wmma_f32_16x16x4_f32(false, a, false, b3, (short)0, acc3, false, false);
    }

    #pragma unroll
    for (int v = 0; v < 8; ++v) {
        const int row = r0 + v + 8 * kh;
        float* hrow = H + (size_t)row * D_OUT + m;   // lanes 0..15 -> contiguous 64B
        hrow[ 0] = acc0[v];
        hrow[16] = acc1[v];
        hrow[32] = acc2[v];
        hrow[48] = acc3[v];
    }
}

// ---------------------------------------------------------------------------
// Kernel 2: out[i, j] = bias[j]  (the "+ b" term; edges accumulate on top)
// ---------------------------------------------------------------------------
__global__ __launch_bounds__(256) void init_out_bias(
    const float* __restrict__ bias, float* __restrict__ out)
{
    const int i = blockIdx.x * blockDim.x + threadIdx.x;
    if (i < N_NODES * D_OUT) out[i] = bias[i & (D_OUT - 1)];
}

// ---------------------------------------------------------------------------
// Kernel 3: edge scatter. 64 threads per edge, one column each.
// Gather of h[src] is a fully-coalesced 256B row read (L2-resident: H is
// 12.8 MB << 192 MB L2); one hardware global_atomic_add_f32 per element.
// ---------------------------------------------------------------------------
__global__ __launch_bounds__(256) void scatter_edges(
    const int*   __restrict__ src,   // [N_EDGES]
    const int*   __restrict__ dst,   // [N_EDGES]
    const float* __restrict__ ew,    // [N_EDGES]
    const float* __restrict__ H,     // [N_NODES, D_OUT]
    float*       __restrict__ out)   // [N_NODES, D_OUT]
{
    const size_t gid = (size_t)blockIdx.x * blockDim.x + threadIdx.x;
    const size_t e = gid >> 6;          // edge id (uniform per 2 waves)
    const int    c = (int)(gid & 63);   // column
    if (e >= N_EDGES) return;

    const int s = src[e];
    const int d = dst[e];
    const float val = ew[e] * H[(size_t)s * D_OUT + c];
    unsafeAtomicAdd(&out[(size_t)d * D_OUT + c], val);  // global_atomic_add_f32
}

extern "C" void kernel_launch(void* const* d_in, const int* in_sizes, int n_in,
                              void* d_out, int out_size, void* d_ws, size_t ws_size,
                              hipStream_t stream) {
    const float* W     = (const float*)d_in[0];   // [64, 256]
    const float* bias  = (const float*)d_in[1];   // [64]
    const int*   edges = (const int*)d_in[2];     // [2, N_EDGES]
    const float* ew    = (const float*)d_in[3];   // [N_EDGES]
    const float* X     = (const float*)d_in[4];   // [50000, 256]
    float* out = (float*)d_out;                   // [50000, 64]
    float* H   = (float*)d_ws;                    // scratch: 50000*64*4 = 12.8 MB

    // 1) H = X @ W^T   (3125 row-tiles, 8 waves/block)
    const int tiles = (N_NODES + 15) / 16;                 // 3125
    gemm_wmma_f32<<<(tiles + 7) / 8, 256, 0, stream>>>(X, W, H);

    // 2) out = bias broadcast
    const int total = N_NODES * D_OUT;                     // 3.2M
    init_out_bias<<<(total + 255) / 256, 256, 0, stream>>>(bias, out);

    // 3) scatter-add edge messages
    const long long work = (long long)N_EDGES * D_OUT;     // 80M lanes
    const int blocks = (int)((work + 255) / 256);          // 312500
    scatter_edges<<<blocks, 256, 0, stream>>>(edges, edges + N_EDGES, ew, X ? H : H, out);
}